// LowRankSig_FirstOrder_54365696032800
// MI455X (gfx1250) — compile-verified
//
#include <hip/hip_runtime.h>
#include <hip/hip_bf16.h>

// LowRankSig depth-4 signature kernel for gfx1250 (MI455X).
//
// dM = dX @ kernel fused with the nested-excumsum scan entirely in LDS
// (avoids 2x168MB of HBM traffic for dM). Time axis is split into SEG
// segments; each segment emits order-<=4 signature transfer coefficients
// (Chen-style), folded by a tiny combine kernel. GEMM uses fp32 WMMA
// v_wmma_f32_16x16x4_f32 to match the fp32 reference.
//
// v2: weight panel stored as packed K-pairs so every WMMA B-operand is a
// single ds_load_b64; scan split across 3 wave-pairs (one per signature
// level); global_prefetch of the next time chunk.

typedef float v2f __attribute__((ext_vector_type(2)));
typedef float v8f __attribute__((ext_vector_type(8)));

#define TT   2048      // time steps
#define BB   32        // batch
#define FIN  63        // raw features
#define FF   64        // features incl. time channel
#define NCOL 640       // 10*64 flattened output columns
#define TC   32        // time chunk per WMMA+scan round
#define SKP  1312      // LDS stride (floats) per K-pair row: 2*640 pad->1312 (mod 64 == 32)
#define SMS  644       // LDS stride for dM tile (pad vs 640)
#define SXS  68        // LDS stride for dX tile (pad vs 64)

__global__ __launch_bounds__(256) void sig_gemm_scan(
    const float* __restrict__ x, const float* __restrict__ ker,
    float* __restrict__ ws, int SEG) {
  extern __shared__ float smem[];
  float* shK = smem;                 // 32 K-pair rows x SKP  (weights, paired)
  float* shX = shK + 32 * SKP;       // TC x SXS  (dX chunk)
  float* shM = shX + TC * SXS;       // TC x SMS  (dM chunk)

  const int tid  = threadIdx.x;
  const int lane = tid & 31;
  const int wave = tid >> 5;
  const int lm   = lane & 15;        // M (A) / N (B,D) within 16-wide tile
  const int lh   = lane >> 4;        // half-wave selector
  const int b    = blockIdx.x / SEG;
  const int sg   = blockIdx.x % SEG;
  const int TS   = TT / SEG;
  const int t0   = sg * TS;
  const float DTC = 2.0f / (float)(TT - 1);   // d(time channel)/dt

  // Stage weights into LDS once, packed as K-pairs:
  // element (kf, col) -> shK[(kf/2)*SKP + col*2 + (kf&1)]
  for (int i = tid; i < FF * NCOL; i += 256) {
    int kf = i / NCOL, c = i - kf * NCOL;
    shK[(kf >> 1) * SKP + c * 2 + (kf & 1)] = ker[i];
  }

  // Per-lane signature state. Waves 0-1: level 1+2 (u=tid);
  // waves 2-3: level 3 (u=tid-64); waves 4-5: level 4 (u=tid-128).
  float accW = 0.f;                                          // level 1
  float sP = 0.f, accQ = 0.f, accPQ = 0.f;                   // level 2
  float sX3 = 0.f, sY = 0.f, sXY = 0.f;                      // level 3
  float accZ = 0.f, accYZ = 0.f, accXYZ = 0.f;
  float sA = 0.f, sB = 0.f, sC = 0.f;                        // level 4
  float sAB = 0.f, sBC = 0.f, sABC = 0.f;
  float accD = 0.f, accCD = 0.f, accBCD = 0.f, accABCD = 0.f;

  const int NCHUNK = TS / TC;
  for (int ch = 0; ch < NCHUNK; ++ch) {
    const int tbase = t0 + ch * TC;
    __syncthreads();   // previous chunk's scan done before overwriting LDS

    // Prefetch next chunk's x rows (32 rows x 63 f32; 8 threads/row).
    {
      int pgt = tbase + TC + (tid >> 3);
      if (pgt < TT)
        __builtin_prefetch(&x[((size_t)b * TT + pgt) * FIN + ((tid & 7) << 3)], 0, 1);
    }

    // ---- Phase A: build dX (32 x 64) = X[t]-X[t-1], row gt==0 is zero ----
    for (int i = tid; i < TC * FF; i += 256) {
      int t = i >> 6, f = i & 63;
      int gt = tbase + t;
      float v;
      if (gt == 0)        v = 0.f;
      else if (f == FIN)  v = DTC;
      else                v = x[((size_t)b * TT + gt) * FIN + f]
                            - x[((size_t)b * TT + gt - 1) * FIN + f];
      shX[t * SXS + f] = v;
    }
    __syncthreads();

    // ---- Phase B: dM(32x640) = dX(32x64) @ K(64x640) via fp32 WMMA ----
    // 2 m-tiles x 40 n-tiles x 16 k-steps; 5 n-tiles per wave.
    for (int nt = wave * 5; nt < wave * 5 + 5; ++nt) {
      const int nb = nt * 16 + lm;
      const float* bcol = &shK[(size_t)lh * SKP + nb * 2];   // kp = 2*kk + lh
      for (int mt = 0; mt < 2; ++mt) {
        v8f acc = {};
        const float* arow = &shX[(mt * 16 + lm) * SXS + 2 * lh];
#pragma unroll
        for (int kk = 0; kk < 16; ++kk) {
          // A vgpr pair holds K = {k0, k0+1}, k0 = 4*kk + 2*lh (8B aligned)
          v2f a  = *(const v2f*)&arow[kk * 4];
          v2f bv = *(const v2f*)&bcol[kk * 2 * SKP];
          acc = __builtin_amdgcn_wmma_f32_16x16x4_f32(
              false, a, false, bv, (short)0, acc, false, false);
        }
        // D layout: lanes 0-15 -> M=j, lanes 16-31 -> M=8+j
#pragma unroll
        for (int j = 0; j < 8; ++j)
          shM[(mt * 16 + 8 * lh + j) * SMS + nb] = acc[j];
      }
    }
    __syncthreads();

    // ---- Phase C: sequential signature scan, one level per wave-pair ----
    if (wave < 2) {                       // level 1 + level 2
      const int u = tid;
      for (int t = 0; t < TC; ++t) {
        const float* r = &shM[t * SMS + u];
        float d0 = r[0], d1 = r[64], d2 = r[128];
        accW += d0;
        accPQ = fmaf(d2, sP, accPQ);  accQ += d2;  sP += d1;
      }
    } else if (wave < 4) {                // level 3
      const int u = tid - 64;
      for (int t = 0; t < TC; ++t) {
        const float* r = &shM[t * SMS + u];
        float d3 = r[192], d4 = r[256], d5 = r[320];
        accXYZ = fmaf(d5, sXY, accXYZ);
        accYZ  = fmaf(d5, sY,  accYZ);   accZ += d5;
        sXY    = fmaf(d4, sX3, sXY);     sY   += d4;  sX3 += d3;
      }
    } else if (wave < 6) {                // level 4
      const int u = tid - 128;
      for (int t = 0; t < TC; ++t) {
        const float* r = &shM[t * SMS + u];
        float d6 = r[384], d7 = r[448], d8 = r[512], d9 = r[576];
        accABCD = fmaf(d9, sABC, accABCD);
        accBCD  = fmaf(d9, sBC,  accBCD);
        accCD   = fmaf(d9, sC,   accCD);   accD += d9;
        sABC    = fmaf(d8, sAB,  sABC);
        sBC     = fmaf(d8, sB,   sBC);     sC   += d8;
        sAB     = fmaf(d7, sA,   sAB);     sB   += d7;
        sA     += d6;
      }
    }
  }

  // ---- Emit per-segment transfer coefficients (20 floats per (b,sg,u)) ----
  if (wave < 2) {
    float* st = ws + (((size_t)b * SEG + sg) * 64 + tid) * 24;
    st[0] = accW; st[1] = sP; st[2] = accQ; st[3] = accPQ;
  } else if (wave < 4) {
    float* st = ws + (((size_t)b * SEG + sg) * 64 + (tid - 64)) * 24;
    st[4] = sX3; st[5] = sY; st[6] = sXY;
    st[7] = accZ; st[8] = accYZ; st[9] = accXYZ;
  } else if (wave < 6) {
    float* st = ws + (((size_t)b * SEG + sg) * 64 + (tid - 128)) * 24;
    st[10] = sA;   st[11] = sB;    st[12] = sC;
    st[13] = sAB;  st[14] = sBC;   st[15] = sABC;
    st[16] = accD; st[17] = accCD; st[18] = accBCD; st[19] = accABCD;
  }
}

// Fold segments in time order (Chen-style combine). grid=B, block=64 (u).
__global__ __launch_bounds__(64) void sig_combine(
    const float* __restrict__ ws, float* __restrict__ out, int SEG) {
  const int b = blockIdx.x;
  const int u = threadIdx.x;
  float s1 = 0.f, s3 = 0.f, s4 = 0.f, s6 = 0.f, s7 = 0.f, s8 = 0.f, Y = 0.f;
  for (int sg = 0; sg < SEG; ++sg) {
    const float* st = ws + (((size_t)b * SEG + sg) * 64 + u) * 24;
    float W = st[0];
    float P = st[1],  Q  = st[2],  PQ = st[3];
    float X = st[4],  Y3 = st[5],  XY = st[6];
    float Z = st[7],  YZ = st[8],  XYZ = st[9];
    float A = st[10], Bv = st[11], C  = st[12];
    float AB = st[13], BC = st[14], ABC = st[15];
    float D = st[16], CD = st[17], BCD = st[18], ABCD = st[19];
    // level 1
    Y += W;
    // level 2
    Y = fmaf(s1, Q, Y + PQ);             s1 += P;
    // level 3
    Y = fmaf(s4, Z, fmaf(s3, YZ, Y + XYZ));
    s4 = fmaf(s3, Y3, s4 + XY);          s3 += X;
    // level 4 (use pre-update s6,s7,s8 throughout)
    Y  = fmaf(s8, D, fmaf(s7, CD, fmaf(s6, BCD, Y + ABCD)));
    s8 = fmaf(s7, C, fmaf(s6, BC, s8 + ABC));
    s7 = fmaf(s6, Bv, s7 + AB);
    s6 += A;
  }
  out[b * 64 + u] = Y;
}

extern "C" void kernel_launch(void* const* d_in, const int* in_sizes, int n_in,
                              void* d_out, int out_size, void* d_ws, size_t ws_size,
                              hipStream_t stream) {
  (void)in_sizes; (void)n_in; (void)out_size;
  const float* x   = (const float*)d_in[0];   // (32,2048,63) f32
  const float* ker = (const float*)d_in[1];   // (64,10,64)   f32
  float* out = (float*)d_out;                 // (32,64)      f32
  float* ws  = (float*)d_ws;

  // Time-segment parallelism sized to available workspace (needs
  // B*SEG*64*24 floats; SEG=8 -> 1.5 MB).
  int SEG = 8;
  while (SEG > 1 && (size_t)BB * SEG * 64 * 24 * sizeof(float) > ws_size)
    SEG >>= 1;

  const size_t shbytes =
      (size_t)(32 * SKP + TC * SXS + TC * SMS) * sizeof(float);  // 259,072 B

  hipLaunchKernelGGL(sig_gemm_scan, dim3(BB * SEG), dim3(256), shbytes, stream,
                     x, ker, ws, SEG);
  hipLaunchKernelGGL(sig_combine, dim3(BB), dim3(64), 0, stream, ws, out, SEG);
}